// MoE_27255862461168
// MI455X (gfx1250) — compile-verified
//
#include <hip/hip_runtime.h>
#include <math.h>

// ---------------------------------------------------------------------------
// MoE SIREN for MI455X (gfx1250).  Compute-bound (157 GFLOP f32 vs ~7 MB
// traffic) -> run all expert GEMMs on V_WMMA_F32_16X16X4_F32 (f32 matrix
// pipe; f16/bf16 would wreck sin(w0*z) phase accuracy with w0 up to 247.5).
// Weight chunk pair-interleaved in LDS (B frag = one ds_load_b64) and placed
// at LDS offset 0 via a manually carved shared buffer, so B-load addresses
// fold into the 16-bit DS immediate instead of per-load v_or_b32.
// ---------------------------------------------------------------------------

typedef __attribute__((ext_vector_type(2))) float v2f;
typedef __attribute__((ext_vector_type(8))) float v8f;

#define N_TOT   65536
#define CC      3
#define GF      16
#define E_NUM   6
#define FDIM    255
#define FP      256     // padded feature dim
#define NP      64      // rows per workgroup
#define KC      32      // K-chunk of weights staged in LDS
#define HIDL    3
#define LN_EPS  1e-5f
#define BW      45.0f

// ===========================================================================
// Expert SIREN kernel: one (expert, 64-row block) per workgroup, 4 waves,
// each wave owns 16 rows and 16 column tiles of 16 (full 256-wide output).
// LDS: weight chunk (32KB, pair-interleaved, at offset 0) + 2x 64x256 f32
// activation ping-pong (64KB each) = 160KB -> exactly 2 WGs per 320KB WGP.
// ===========================================================================
__global__ __launch_bounds__(128) void moe_expert_kernel(
    const float* __restrict__ coords,
    const float* __restrict__ Wf,  const float* __restrict__ bf,
    const float* __restrict__ We0, const float* __restrict__ be0,
    const float* __restrict__ Weh, const float* __restrict__ beh,
    const float* __restrict__ Weo, const float* __restrict__ beo,
    float* __restrict__ xws)                 // [N, E] row-major
{
    // Manual LDS layout: wS at offset 0 (hot B-loads get small DS immediates),
    // then the two activation buffers.
    __shared__ __align__(16) unsigned char ldsbuf[(KC * FP + 2 * NP * FP) * 4];
    float (*wS)[FP][2] = reinterpret_cast<float (*)[FP][2]>(ldsbuf);            // 32 KB
    float (*yA)[FP]    = reinterpret_cast<float (*)[FP]>(ldsbuf + KC * FP * 4); // 64 KB
    float (*yB)[FP]    = reinterpret_cast<float (*)[FP]>(ldsbuf + (KC + NP) * FP * 4);

    const int e    = blockIdx.x;
    const int row0 = blockIdx.y * NP;
    const int t    = threadIdx.x;            // 0..127
    const int lane = t & 31;
    const int hs   = lane >> 4;              // half-wave select
    const int mm   = lane & 15;
    const int r0   = (t >> 5) << 4;          // wave's 16-row base in tile
    const float w0 = 0.5f * BW + BW * (float)e;

    // ---- shared feature = coords @ Wf + bf  (64 rows x 16 cols) ----
    for (int i = t; i < NP * GF; i += 128) {
        int r = i >> 4, c = i & 15;
        const float* crd = coords + (row0 + r) * CC;
        yA[r][c] = bf[c] + crd[0] * Wf[c] + crd[1] * Wf[GF + c] + crd[2] * Wf[2 * GF + c];
    }
    // ---- stage W0 (16 x 255, zero-padded, pair-interleaved) ----
    for (int i = t; i < GF * FP; i += 128) {
        int kk = i >> 8, n = i & 255;
        wS[kk >> 1][n][kk & 1] = (n < FDIM) ? We0[(e * GF + kk) * FDIM + n] : 0.0f;
    }
    __syncthreads();

    v8f acc[16];

    // ================= layer 0: y = sin(w0*(feature @ W0 + b0)) =============
    #pragma unroll
    for (int ct = 0; ct < 16; ++ct) acc[ct] = {};
    #pragma unroll
    for (int k = 0; k < GF; k += 4) {
        // A 16x4: lanes 0-15 hold K=k..k+1, lanes 16-31 hold K=k+2..k+3
        v2f a = *reinterpret_cast<const v2f*>(&yA[r0 + mm][k + 2 * hs]);
        #pragma unroll
        for (int ct = 0; ct < 16; ++ct) {
            v2f b = *reinterpret_cast<const v2f*>(&wS[(k >> 1) + hs][ct * 16 + mm][0]);
            acc[ct] = __builtin_amdgcn_wmma_f32_16x16x4_f32(
                false, a, false, b, (short)0, acc[ct], false, false);
        }
    }
    #pragma unroll
    for (int ct = 0; ct < 16; ++ct) {
        int col = ct * 16 + mm;
        float bias = (col < FDIM) ? be0[e * FDIM + col] : 0.0f;
        #pragma unroll
        for (int v = 0; v < 8; ++v)          // D: row = r0 + 8*hs + v
            yB[r0 + 8 * hs + v][col] = sinf(w0 * (acc[ct][v] + bias));
    }
    __syncthreads();

    // ================= hidden layers: y = sin(w0*(y @ Wh + bh)) =============
    float (*src)[FP] = yB;
    float (*dst)[FP] = yA;
    for (int l = 0; l < HIDL; ++l) {
        #pragma unroll
        for (int ct = 0; ct < 16; ++ct) acc[ct] = {};
        const float* Wl = Weh + (e * HIDL + l) * FDIM * FDIM;
        for (int kc = 0; kc < FP; kc += KC) {
            for (int i = t; i < KC * FP; i += 128) {          // stage K-chunk
                int kk = i >> 8, n = i & 255;
                int krow = kc + kk;
                wS[kk >> 1][n][kk & 1] =
                    (krow < FDIM && n < FDIM) ? Wl[krow * FDIM + n] : 0.0f;
            }
            __syncthreads();
            #pragma unroll
            for (int k = 0; k < KC; k += 4) {
                v2f a = *reinterpret_cast<const v2f*>(&src[r0 + mm][kc + k + 2 * hs]);
                #pragma unroll
                for (int ct = 0; ct < 16; ++ct) {
                    v2f b = *reinterpret_cast<const v2f*>(
                        &wS[(k >> 1) + hs][ct * 16 + mm][0]);
                    acc[ct] = __builtin_amdgcn_wmma_f32_16x16x4_f32(
                        false, a, false, b, (short)0, acc[ct], false, false);
                }
            }
            __syncthreads();
        }
        const float* bh = beh + (e * HIDL + l) * FDIM;
        #pragma unroll
        for (int ct = 0; ct < 16; ++ct) {
            int col = ct * 16 + mm;
            float bias = (col < FDIM) ? bh[col] : 0.0f;
            #pragma unroll
            for (int v = 0; v < 8; ++v)
                dst[r0 + 8 * hs + v][col] = sinf(w0 * (acc[ct][v] + bias));
        }
        __syncthreads();
        float (*tmp)[FP] = src; src = dst; dst = tmp;
    }
    // final activations are in `src`

    // ================= output head: x = y @ Wo + bo =========================
    const float* Wo = Weo + e * FDIM;
    const int rr = t & 63, half = t >> 6;
    float partial = 0.0f;
    for (int j = half * 128; j < half * 128 + 128; ++j)
        if (j < FDIM) partial += src[rr][j] * Wo[j];
    float* red = &wS[0][0][0];               // reuse weight-chunk LDS
    red[t] = partial;
    __syncthreads();
    if (t < 64)
        xws[(row0 + t) * E_NUM + e] = red[t] + red[t + 64] + beo[e];
}

// ===========================================================================
// Gate kernel: tiny per-row scalar MLP; only needed for m[e] = sum softmax
// (the KLD loss).  ~67 MFLOP total, negligible.
// ===========================================================================
__global__ __launch_bounds__(256) void moe_gate_kernel(
    const float* __restrict__ coords,
    const float* __restrict__ Wf,  const float* __restrict__ bf,
    const float* __restrict__ gW0, const float* __restrict__ gb0,
    const float* __restrict__ gW1, const float* __restrict__ gb1,
    const float* __restrict__ gW2, const float* __restrict__ gb2,
    const float* __restrict__ ln_g, const float* __restrict__ ln_b,
    const float* __restrict__ gW3, const float* __restrict__ gb3,
    float* __restrict__ m)                   // [E] expert load sums
{
    __shared__ float sm[E_NUM];
    if (threadIdx.x < E_NUM) sm[threadIdx.x] = 0.0f;
    __syncthreads();

    const int n = blockIdx.x * blockDim.x + threadIdx.x;
    float p[E_NUM];
    if (n < N_TOT) {
        float f[GF], h[GF], h2[GF];
        const float c0 = coords[n * 3], c1 = coords[n * 3 + 1], c2 = coords[n * 3 + 2];
        #pragma unroll
        for (int j = 0; j < GF; ++j)
            f[j] = bf[j] + c0 * Wf[j] + c1 * Wf[GF + j] + c2 * Wf[2 * GF + j];
        #pragma unroll
        for (int j = 0; j < GF; ++j) {
            float s = gb0[j];
            #pragma unroll
            for (int i = 0; i < GF; ++i) s += f[i] * gW0[i * GF + j];
            h[j] = fmaxf(s, 0.0f);
        }
        #pragma unroll
        for (int j = 0; j < GF; ++j) {
            float s = gb1[j];
            #pragma unroll
            for (int i = 0; i < GF; ++i) s += h[i] * gW1[i * GF + j];
            h2[j] = fmaxf(s, 0.0f);
        }
        #pragma unroll
        for (int j = 0; j < GF; ++j) {
            float s = gb2[j];
            #pragma unroll
            for (int i = 0; i < GF; ++i) s += h2[i] * gW2[i * GF + j];
            h[j] = s;
        }
        float mu = 0.0f;
        #pragma unroll
        for (int j = 0; j < GF; ++j) mu += h[j];
        mu *= (1.0f / GF);
        float var = 0.0f;
        #pragma unroll
        for (int j = 0; j < GF; ++j) { float d = h[j] - mu; var += d * d; }
        var *= (1.0f / GF);
        const float inv = rsqrtf(var + LN_EPS);
        #pragma unroll
        for (int j = 0; j < GF; ++j) h[j] = (h[j] - mu) * inv * ln_g[j] + ln_b[j];

        float lg[E_NUM], mx = -1e30f;
        #pragma unroll
        for (int ee = 0; ee < E_NUM; ++ee) {
            float s = gb3[ee];
            #pragma unroll
            for (int i = 0; i < GF; ++i) s += h[i] * gW3[i * E_NUM + ee];
            lg[ee] = s; mx = fmaxf(mx, s);
        }
        float se = 0.0f;
        #pragma unroll
        for (int ee = 0; ee < E_NUM; ++ee) { p[ee] = expf(lg[ee] - mx); se += p[ee]; }
        const float rse = 1.0f / se;
        #pragma unroll
        for (int ee = 0; ee < E_NUM; ++ee) {
            p[ee] *= rse;
            atomicAdd(&sm[ee], p[ee]);
        }
    }
    __syncthreads();
    if (threadIdx.x < E_NUM) atomicAdd(&m[threadIdx.x], sm[threadIdx.x]);
}

// ===========================================================================
// Finalize: TOPK=1 masked_scatter_ semantics collapse to
// output[n] = x.flatten()[n]; thread 0 computes the KLD load-balance loss.
// ===========================================================================
__global__ __launch_bounds__(256) void moe_finalize_kernel(
    const float* __restrict__ xws, const float* __restrict__ m,
    float* __restrict__ out)                 // [N] output then [1] kld
{
    const int n = blockIdx.x * blockDim.x + threadIdx.x;
    if (n < N_TOT) out[n] = xws[n];
    if (n == 0) {
        const float q = (float)N_TOT / (float)E_NUM;
        float kld = 0.0f;
        for (int ee = 0; ee < E_NUM; ++ee)
            kld += m[ee] * (logf(m[ee]) - logf(q));
        out[N_TOT] = kld / (float)N_TOT / (float)E_NUM;
    }
}

// ===========================================================================
extern "C" void kernel_launch(void* const* d_in, const int* in_sizes, int n_in,
                              void* d_out, int out_size, void* d_ws, size_t ws_size,
                              hipStream_t stream)
{
    const float* coords = (const float*)d_in[0];
    const float* Wf   = (const float*)d_in[1];
    const float* bf   = (const float*)d_in[2];
    const float* gW0  = (const float*)d_in[3];
    const float* gb0  = (const float*)d_in[4];
    const float* gW1  = (const float*)d_in[5];
    const float* gb1  = (const float*)d_in[6];
    const float* gW2  = (const float*)d_in[7];
    const float* gb2  = (const float*)d_in[8];
    const float* ln_g = (const float*)d_in[9];
    const float* ln_b = (const float*)d_in[10];
    const float* gW3  = (const float*)d_in[11];
    const float* gb3  = (const float*)d_in[12];
    const float* We0  = (const float*)d_in[13];
    const float* be0  = (const float*)d_in[14];
    const float* Weh  = (const float*)d_in[15];
    const float* beh  = (const float*)d_in[16];
    const float* Weo  = (const float*)d_in[17];
    const float* beo  = (const float*)d_in[18];

    float* xws = (float*)d_ws;                         // [N, E] expert outputs
    float* m   = (float*)((char*)d_ws + (size_t)N_TOT * E_NUM * sizeof(float));
    float* out = (float*)d_out;                        // [N] + [1] kld

    hipMemsetAsync(m, 0, E_NUM * sizeof(float), stream);

    moe_gate_kernel<<<N_TOT / 256, 256, 0, stream>>>(
        coords, Wf, bf, gW0, gb0, gW1, gb1, gW2, gb2, ln_g, ln_b, gW3, gb3, m);

    moe_expert_kernel<<<dim3(E_NUM, N_TOT / NP), 128, 0, stream>>>(
        coords, Wf, bf, We0, be0, Weh, beh, Weo, beo, xws);

    moe_finalize_kernel<<<N_TOT / 256, 256, 0, stream>>>(xws, m, out);
}